// hybridKinematicSim_26414048870687
// MI455X (gfx1250) — compile-verified
//
#include <hip/hip_runtime.h>

typedef __attribute__((ext_vector_type(2))) float v2f;
typedef __attribute__((ext_vector_type(8))) float v8f;

#define DT_F    0.01f
#define PI_F    3.14159265358979323846f
#define INV2PI  0.15915494309189535f

__device__ __forceinline__ float fast_tanh(float x) {
    x = fminf(10.0f, fmaxf(-10.0f, x));
    float e = __expf(2.0f * x);
    return (e - 1.0f) * __builtin_amdgcn_rcpf(e + 1.0f);
}

__global__ __launch_bounds__(256) void hybrid_rollout_kernel(
    const float* __restrict__ fs,    // (B,8,8)
    const float* __restrict__ act,   // (B,50,2)
    const float* __restrict__ W1,    // (16,40)
    const float* __restrict__ b1,    // (16)
    const float* __restrict__ W2,    // (3,16)
    const float* __restrict__ b2,    // (3)
    const float* __restrict__ lf_p,
    const float* __restrict__ lr_p,
    const float* __restrict__ toff_p,
    const float* __restrict__ trat_p,
    const float* __restrict__ uc_p,
    const float* __restrict__ rb_p,  // (3)
    const int*   __restrict__ en_p,
    float* __restrict__ out,         // (B,50,8)
    int B)
{
    const int tid  = threadIdx.x;
    const int lane = tid & 31;
    const int half = lane >> 4;       // 0: lanes 0-15, 1: lanes 16-31
    const int n    = lane & 15;       // batch slot / matrix row
    const int wave = tid >> 5;

    long b = (long)blockIdx.x * 128 + wave * 16 + n;
    const bool live = (b < B);
    if (b >= B) b = B - 1;            // clamp; keeps EXEC all-ones for WMMA

    const float lf   = *lf_p, lr = *lr_p;
    const float toff = *toff_p, trat = *trat_p, uc = *uc_p;
    const float rb0 = rb_p[0], rb1 = rb_p[1], rb2 = rb_p[2];
    const float b20 = b2[0], b21 = b2[1], b22 = b2[2];
    const float en   = (*en_p != 0) ? 1.0f : 0.0f;
    const float invL = 1.0f / (lf + lr);

    // ---- W1 into WMMA A-layout registers: 10 chunks of K=4 -----------------
    // A 16x4 f32 layout: lane m (0-15) holds K=4c+0 (v.x), 4c+1 (v.y);
    // lane m+16 holds K=4c+2, 4c+3.
    v2f A[10];
#pragma unroll
    for (int c = 0; c < 10; ++c) {
        const float* p = W1 + n * 40 + 4 * c + 2 * half;
        A[c].x = p[0];
        A[c].y = p[1];
    }
    // bias in C/D layout: VGPR r -> row M = r + 8*half
    float bias[8];
#pragma unroll
    for (int r = 0; r < 8; ++r) bias[r] = b1[r + 8 * half];
    // W2 slice for this lane-half's h[0..7] / h[8..15]
    float w2r[3][8];
#pragma unroll
    for (int j = 0; j < 3; ++j)
#pragma unroll
        for (int r = 0; r < 8; ++r) w2r[j][r] = W2[j * 16 + 8 * half + r];

    // ---- build initial feature window in WMMA B-layout registers -----------
    // Feature index K = f*8 + t (f in 0..4, t in 0..7).
    // B 4x16 f32 layout: chunk c: v.x = {low:K=4c, high:K=4c+2}, v.y = {low:4c+1, high:4c+3}
    v2f Bm[10];
#pragma unroll
    for (int c = 0; c < 10; ++c) { Bm[c].x = 0.0f; Bm[c].y = 0.0f; }

    const float* fsb = fs + b * 64;
    float x_, y_, vx_, vy_, psi_, om_, th_, st_, cp = 1.0f, sp = 0.0f;
    x_ = y_ = vx_ = vy_ = psi_ = om_ = th_ = st_ = 0.0f;

#pragma unroll
    for (int t = 0; t < 8; ++t) {
        float hx  = fsb[t * 8 + 0], hvx = fsb[t * 8 + 1];
        float hy  = fsb[t * 8 + 2], hvy = fsb[t * 8 + 3];
        float hps = fsb[t * 8 + 4], hom = fsb[t * 8 + 5];
        float hth = fsb[t * 8 + 6], hst = fsb[t * 8 + 7];
        float c_ = __cosf(hps), s_ = __sinf(hps);
        float feat[5];
        feat[0] =  hvx * c_ + hvy * s_;   // Vx_h
        feat[1] = -hvx * s_ + hvy * c_;   // Vy_h
        feat[2] = hom; feat[3] = hth; feat[4] = hst;
        const int  cb   = t >> 2;               // chunk offset within feature block
        const int  j    = t & 1;                // .x or .y
        const bool mine = (((t >> 1) & 1) == half);
#pragma unroll
        for (int f = 0; f < 5; ++f) {
            int c = 2 * f + cb;
            if (j == 0) Bm[c].x = mine ? feat[f] : Bm[c].x;
            else        Bm[c].y = mine ? feat[f] : Bm[c].y;
        }
        if (t == 7) {
            x_ = hx; vx_ = hvx; y_ = hy; vy_ = hvy;
            psi_ = hps; om_ = hom; th_ = hth; st_ = hst;
            cp = c_; sp = s_;
        }
    }

    const float* ab = act + b * 100;
    float* ob = out + b * 400;

#pragma unroll 1
    for (int i = 0; i < 50; ++i) {
        // ---- layer 1: h = W1 @ x + b1 via 10x V_WMMA_F32_16X16X4_F32 ------
        v8f acc0, acc1;
#pragma unroll
        for (int r = 0; r < 8; ++r) { acc0[r] = bias[r]; acc1[r] = 0.0f; }
#pragma unroll
        for (int c = 0; c < 10; c += 2) {
            acc0 = __builtin_amdgcn_wmma_f32_16x16x4_f32(
                false, A[c],     false, Bm[c],     (short)0, acc0, false, false);
            acc1 = __builtin_amdgcn_wmma_f32_16x16x4_f32(
                false, A[c + 1], false, Bm[c + 1], (short)0, acc1, false, false);
        }
        float h[8];
#pragma unroll
        for (int r = 0; r < 8; ++r) h[r] = fast_tanh(acc0[r] + acc1[r]);

        // ---- layer 2: partial dot per half, reduce across lane pair -------
        float p0 = 0.0f, p1 = 0.0f, p2 = 0.0f;
#pragma unroll
        for (int r = 0; r < 8; ++r) {
            p0 = fmaf(w2r[0][r], h[r], p0);
            p1 = fmaf(w2r[1][r], h[r], p1);
            p2 = fmaf(w2r[2][r], h[r], p2);
        }
        p0 += __shfl_xor(p0, 16, 32);
        p1 += __shfl_xor(p1, 16, 32);
        p2 += __shfl_xor(p2, 16, 32);
        float y0 = fast_tanh(p0 + b20) * rb0 * en;
        float y1 = fast_tanh(p1 + b21) * rb1 * en;
        float y2 = fast_tanh(p2 + b22) * rb2 * en;

        // ---- kinematic step (duplicated on both lane halves) --------------
        float Vx = vx_ * cp + vy_ * sp;
        float t_ = __tanf(st_);
        float zt = t_ * invL;                 // tan(steer)/L  (= 1/R)
        float z  = lr * zt;                   // lr/R
        float sinb = z * __builtin_amdgcn_rsqf(1.0f + z * z); // sin(atan(z))
        Vx += (th_ - toff) * trat * DT_F;
        float Vy = __builtin_amdgcn_sqrtf(vx_ * vx_ + vy_ * vy_) * sinb;
        float om = Vx * zt;
        Vx += y0 * DT_F; Vy += y1 * DT_F; om += y2 * DT_F;
        Vy -= uc * Vx * st_;
        float Vxg = Vx * cp - Vy * sp;
        float Vyg = Vx * sp + Vy * cp;
        float pm  = psi_ + om * DT_F + PI_F;
        float psin = pm - 2.0f * PI_F * floorf(pm * INV2PI) - PI_F;
        float xn = x_ + Vxg * DT_F;
        float yn = y_ + Vyg * DT_F;
        float a0 = ab[2 * i], a1 = ab[2 * i + 1];

        if (live && lane < 16) {
            float4* o = (float4*)(ob + i * 8);
            o[0] = make_float4(xn, Vxg, yn, Vyg);
            o[1] = make_float4(psin, om, a0, a1);
        }

        // ---- new feature entry (rotation by the entry's own psi) ----------
        float cn = __cosf(psin), sn = __sinf(psin);
        float nf[5];
        nf[0] =  Vxg * cn + Vyg * sn;
        nf[1] = -Vxg * sn + Vyg * cn;
        nf[2] = om; nf[3] = a0; nf[4] = a1;

        // ---- shift feature window one step in K (register-resident) -------
        float S[10];
#pragma unroll
        for (int c = 0; c < 10; ++c) S[c] = __shfl_xor(Bm[c].x, 16, 32);
#pragma unroll
        for (int f = 0; f < 5; ++f) {
            const int c0 = 2 * f, c1 = 2 * f + 1;
            Bm[c0].x = Bm[c0].y;
            Bm[c0].y = (half == 0) ? S[c0] : S[c1];
            Bm[c1].x = Bm[c1].y;
            Bm[c1].y = (half == 0) ? S[c1] : nf[f];
        }

        // ---- advance state -------------------------------------------------
        x_ = xn; y_ = yn; vx_ = Vxg; vy_ = Vyg;
        psi_ = psin; om_ = om; th_ = a0; st_ = a1;
        cp = cn; sp = sn;
    }
}

extern "C" void kernel_launch(void* const* d_in, const int* in_sizes, int n_in,
                              void* d_out, int out_size, void* d_ws, size_t ws_size,
                              hipStream_t stream) {
    const float* fs   = (const float*)d_in[0];   // full_states (B,8,8)
    const float* act  = (const float*)d_in[1];   // actions (B,50,2)
    const float* W1   = (const float*)d_in[2];   // (16,40)
    const float* b1   = (const float*)d_in[3];   // (16)
    const float* W2   = (const float*)d_in[4];   // (3,16)
    const float* b2   = (const float*)d_in[5];   // (3)
    const float* lf   = (const float*)d_in[6];
    const float* lr   = (const float*)d_in[7];
    const float* toff = (const float*)d_in[8];
    const float* trat = (const float*)d_in[9];
    const float* uc   = (const float*)d_in[10];
    const float* rb   = (const float*)d_in[11];  // (3)
    const int*   en   = (const int*)d_in[12];

    const int B = in_sizes[0] / 64;              // (B,8,8)
    const int blocks = (B + 127) / 128;          // 128 batches per 256-thread block

    hybrid_rollout_kernel<<<blocks, 256, 0, stream>>>(
        fs, act, W1, b1, W2, b2, lf, lr, toff, trat, uc, rb, en,
        (float*)d_out, B);
}